// HardNegativeContrastive_50869592655659
// MI455X (gfx1250) — compile-verified
//
#include <hip/hip_runtime.h>
#include <hip/hip_bf16.h>
#include <math.h>

#define B_DIM 4096
#define D_DIM 256
#define C_DIM 50000
#define K_NEG 5
#define LDS_STRIDE 260   // 256 + 4 pad: lane bank = k + 4*row -> conflict-free

typedef __attribute__((ext_vector_type(2))) float v2f;
typedef __attribute__((ext_vector_type(4))) float v4f;
typedef __attribute__((ext_vector_type(8))) float v8f;

// ---------------------------------------------------------------------------
// Kernel 1: per-row top-K (K=5) over teacher_logits with true-label masked.
// One 256-thread block per row. Streaming scan uses non-temporal b128 loads
// (819 MB read-once >> 192 MB L2; keep L2 for prototypes/W/X). Register
// top-5 kept sorted desc via static compare-and-swap insertion (no dynamic
// indexing -> no scratch), then LDS tree merge.
// ---------------------------------------------------------------------------
__global__ __launch_bounds__(256) void topk_kernel(
    const float* __restrict__ logits, const int* __restrict__ labels,
    int* __restrict__ neg_idx) {
  const int row = blockIdx.x;
  const int tid = threadIdx.x;
  // row pitch = 200000 B (16B aligned), C_DIM = 4 * 12500 exactly
  const v4f* rowp4 = (const v4f*)(logits + (size_t)row * C_DIM);
  const int lbl = labels[row];

  float v[K_NEG];
  int   ix[K_NEG];
#pragma unroll
  for (int i = 0; i < K_NEG; i++) { v[i] = -INFINITY; ix[i] = 0; }

  const int C4 = C_DIM / 4;   // 12500
  for (int c4 = tid; c4 < C4; c4 += 256) {
    v4f x = __builtin_nontemporal_load(rowp4 + c4);   // global_load_b128 NT
    const int cbase = c4 * 4;
#pragma unroll
    for (int j = 0; j < 4; j++) {
      const float xv = x[j];
      const int c = cbase + j;
      if (c == lbl) continue;
      if (xv > v[K_NEG - 1]) {
        float cv = xv; int ci = c;
#pragma unroll
        for (int i = 0; i < K_NEG; i++) {
          if (cv > v[i]) {
            float tv = v[i]; int ti = ix[i];
            v[i] = cv; ix[i] = ci;
            cv = tv;  ci = ti;
          }
        }
      }
    }
  }

  __shared__ float sv[256 * K_NEG];
  __shared__ int   si[256 * K_NEG];

  for (int s = 128; s >= 1; s >>= 1) {
    if (tid >= s && tid < 2 * s) {
#pragma unroll
      for (int i = 0; i < K_NEG; i++) { sv[tid * K_NEG + i] = v[i]; si[tid * K_NEG + i] = ix[i]; }
    }
    __syncthreads();
    if (tid < s) {
#pragma unroll
      for (int i = 0; i < K_NEG; i++) {
        float cv = sv[(tid + s) * K_NEG + i];
        int   ci = si[(tid + s) * K_NEG + i];
        if (cv > v[K_NEG - 1]) {
#pragma unroll
          for (int j = 0; j < K_NEG; j++) {
            if (cv > v[j]) {
              float tv = v[j]; int ti = ix[j];
              v[j] = cv; ix[j] = ci;
              cv = tv;  ci = ti;
            }
          }
        }
      }
    }
    __syncthreads();
  }

  if (tid == 0) {
#pragma unroll
    for (int i = 0; i < K_NEG; i++) neg_idx[row * K_NEG + i] = ix[i];
  }
}

// ---------------------------------------------------------------------------
// Kernel 2: fused projector (Linear->ReLU->Linear via V_WMMA_F32_16X16X4_F32)
// + both InfoNCE per-row losses. Block = 16 rows x 256 cols, 16 waves.
// ---------------------------------------------------------------------------
__global__ __launch_bounds__(512) void fused_proj_loss_kernel(
    const float* __restrict__ X, const int* __restrict__ labels,
    const float* __restrict__ prototypes,
    const float* __restrict__ W1, const float* __restrict__ b1,
    const float* __restrict__ W2, const float* __restrict__ b2,
    const int* __restrict__ neg_idx, float* __restrict__ row_loss) {
  __shared__ float hbuf[16 * LDS_STRIDE];
  __shared__ float zbuf[16 * LDS_STRIDE];

  const int tid  = threadIdx.x;
  const int wave = tid >> 5;      // 0..15
  const int lane = tid & 31;
  const int l16  = lane & 15;     // col within tile / row within tile
  const int hi   = lane >> 4;     // lane half -> K offset 2*hi
  const int m0   = blockIdx.x * 16;
  const int n0   = wave * 16;

  // ---- Phase A: h = relu(X @ W1^T + b1), 16x16 tile per wave --------------
  // A frag (16x4 f32): VGPR j, lane-half g: element (M=l16, K=k0+2g+j)
  // B frag (4x16 f32): VGPR j, lane-half g: element (K=k0+2g+j, N=l16)
  //   B[k][n] = W1[n][k]  (torch Linear: x @ W1.T)
  v8f acc = {};
  const float* xrow = X  + (size_t)(m0 + l16) * D_DIM;
  const float* wrow = W1 + (size_t)(n0 + l16) * D_DIM;
  for (int k0 = 0; k0 < D_DIM; k0 += 4) {
    const int kk = k0 + 2 * hi;
    v2f a = *(const v2f*)(xrow + kk);
    v2f b = *(const v2f*)(wrow + kk);
    acc = __builtin_amdgcn_wmma_f32_16x16x4_f32(
        false, a, false, b, (short)0, acc, false, false);
  }
  {
    const float bias = b1[n0 + l16];
#pragma unroll
    for (int i = 0; i < 8; i++) {
      const int m = i + 8 * hi;   // C/D layout: VGPR i holds rows i and i+8
      float hv = acc[i] + bias;
      hbuf[m * LDS_STRIDE + n0 + l16] = hv > 0.f ? hv : 0.f;
    }
  }
  __syncthreads();

  // ---- Phase B: z = h @ W2^T + b2, A fragments from LDS (ds_load_b64) -----
  v8f acc2 = {};
  const float* w2row = W2 + (size_t)(n0 + l16) * D_DIM;
  const float* hrow  = hbuf + l16 * LDS_STRIDE;
  for (int k0 = 0; k0 < D_DIM; k0 += 4) {
    const int kk = k0 + 2 * hi;
    v2f a = *(const v2f*)(hrow + kk);        // 8B-aligned LDS read
    v2f b = *(const v2f*)(w2row + kk);
    acc2 = __builtin_amdgcn_wmma_f32_16x16x4_f32(
        false, a, false, b, (short)0, acc2, false, false);
  }
  {
    const float bias = b2[n0 + l16];
#pragma unroll
    for (int i = 0; i < 8; i++) {
      const int m = i + 8 * hi;
      zbuf[m * LDS_STRIDE + n0 + l16] = acc2[i] + bias;
    }
  }
  __syncthreads();

  // ---- Phase C: InfoNCE losses; wave w owns global row m0 + w -------------
  const int r   = m0 + wave;
  const int lbl = labels[r];
  int nidx[K_NEG];
#pragma unroll
  for (int k = 0; k < K_NEG; k++) nidx[k] = neg_idx[r * K_NEG + k];

  const float* zrow = zbuf + wave * LDS_STRIDE;
  const float* cpos = prototypes + (size_t)lbl * D_DIM;

  float pd = 0.f;
  float n1[K_NEG] = {0.f, 0.f, 0.f, 0.f, 0.f};
  float n2[K_NEG] = {0.f, 0.f, 0.f, 0.f, 0.f};
  for (int d = lane; d < D_DIM; d += 32) {
    const float zv = zrow[d];
    const float cp = cpos[d];
    pd += zv * cp;
#pragma unroll
    for (int k = 0; k < K_NEG; k++) {
      const float cn = prototypes[(size_t)nidx[k] * D_DIM + d];
      n1[k] += zv * cn;
      n2[k] += cp * cn;
    }
  }
  // wave32 tree reduction
#pragma unroll
  for (int off = 16; off >= 1; off >>= 1) {
    pd += __shfl_xor(pd, off, 32);
#pragma unroll
    for (int k = 0; k < K_NEG; k++) {
      n1[k] += __shfl_xor(n1[k], off, 32);
      n2[k] += __shfl_xor(n2[k], off, 32);
    }
  }

  if (lane == 0) {
    const float invT = 1.0f / 0.07f;
    const float pos = pd * invT;

    float m1 = pos, m2 = pos;
#pragma unroll
    for (int k = 0; k < K_NEG; k++) {
      const float s1 = n1[k] * invT; if (s1 > m1) m1 = s1;
      const float s2 = n2[k] * invT; if (s2 > m2) m2 = s2;
    }
    float se1 = expf(pos - m1), se2 = expf(pos - m2);
#pragma unroll
    for (int k = 0; k < K_NEG; k++) {
      se1 += expf(n1[k] * invT - m1);
      se2 += expf(n2[k] * invT - m2);
    }
    const float l1 = logf(se1) + m1 - pos;   // lse - pos = -log_softmax[0]
    const float l2 = logf(se2) + m2 - pos;
    row_loss[r] = l1 + l2;
  }
}

// ---------------------------------------------------------------------------
// Kernel 3: deterministic reduction of 4096 per-row losses -> scalar
// ---------------------------------------------------------------------------
__global__ __launch_bounds__(256) void reduce_kernel(
    const float* __restrict__ row_loss, float* __restrict__ out) {
  __shared__ float s[256];
  const int tid = threadIdx.x;
  float acc = 0.f;
  for (int i = tid; i < B_DIM; i += 256) acc += row_loss[i];
  s[tid] = acc;
  __syncthreads();
  for (int st = 128; st >= 1; st >>= 1) {
    if (tid < st) s[tid] += s[tid + st];
    __syncthreads();
  }
  if (tid == 0) out[0] = s[0] / (float)B_DIM;
}

// ---------------------------------------------------------------------------
extern "C" void kernel_launch(void* const* d_in, const int* in_sizes, int n_in,
                              void* d_out, int out_size, void* d_ws, size_t ws_size,
                              hipStream_t stream) {
  const float* X          = (const float*)d_in[0];  // (B, D)
  const int*   labels     = (const int*)  d_in[1];  // (B,)
  const float* tlogits    = (const float*)d_in[2];  // (B, C)
  const float* prototypes = (const float*)d_in[3];  // (C, D)
  const float* W1         = (const float*)d_in[4];  // (D, D)
  const float* b1         = (const float*)d_in[5];  // (D,)
  const float* W2         = (const float*)d_in[6];  // (D, D)
  const float* b2         = (const float*)d_in[7];  // (D,)

  int*   neg      = (int*)d_ws;                                    // B*K ints
  float* row_loss = (float*)((char*)d_ws + (size_t)B_DIM * K_NEG * sizeof(int));

  topk_kernel<<<B_DIM, 256, 0, stream>>>(tlogits, labels, neg);
  fused_proj_loss_kernel<<<B_DIM / 16, 512, 0, stream>>>(
      X, labels, prototypes, W1, b1, W2, b2, neg, row_loss);
  reduce_kernel<<<1, 256, 0, stream>>>(row_loss, (float*)d_out);
}